// RTX3060GaussianRenderer_52338471469396
// MI455X (gfx1250) — compile-verified
//
#include <hip/hip_runtime.h>
#include <stdint.h>
#include <math.h>

#define N_GAUSS 4096
#define IMG_H 512
#define IMG_W 512
#define RCHUNK 1024
// -0.5 * log2(e): folded into inv_var so alpha = amp * exp2(dx^2*ivxs + dy^2*ivys)
#define NEG_HALF_LOG2E (-0.7213475204444817f)

typedef float v2f __attribute__((ext_vector_type(2)));
typedef float v8f __attribute__((ext_vector_type(8)));
typedef unsigned int u32x4 __attribute__((ext_vector_type(4)));
typedef int i32x4 __attribute__((ext_vector_type(4)));
typedef int i32x8 __attribute__((ext_vector_type(8)));

__device__ __forceinline__ float fast_exp2(float x) {
#if __has_builtin(__builtin_amdgcn_exp2f)
    return __builtin_amdgcn_exp2f(x);   // raw v_exp_f32
#else
    return exp2f(x);
#endif
}

// Low 32 bits of a generic pointer into LDS == LDS byte offset (flat->LDS
// addressing truncates the aperture bits; addrspacecast puts the AS3 offset
// in the low dword).
__device__ __forceinline__ uint32_t lds_byte_addr(const void* p) {
    return (uint32_t)(uintptr_t)p;
}

#if defined(__HIP_DEVICE_COMPILE__) && __has_builtin(__builtin_amdgcn_tensor_load_to_lds)
#define HAVE_TDM 1
// One-instruction DMA of n_dwords*4 bytes (1-D tile) from global to LDS.
// D# layout per CDNA5 ISA ch.8: group0 = count/lds_addr/global_addr/type,
// group1 = data_size + tensor/tile dims + strides. Tracked with TENSORcnt.
// This toolchain exposes the 6-arg builtin: (v4u, v8i, v4i, v4i, v8i, i32 cpol);
// the trailing groups are unused for a count=1, 1-D descriptor -> zeros.
__device__ __forceinline__ void tdm_load_chunk(const void* gsrc,
                                               uint32_t lds_addr,
                                               uint32_t n_dwords) {
    uint64_t ga = (uint64_t)(uintptr_t)gsrc;
    u32x4 g0;
    g0[0] = 1u;                                                // count=1 (valid), user mode
    g0[1] = lds_addr;                                          // lds_addr (bytes)
    g0[2] = (uint32_t)ga;                                      // global_addr[31:0]
    g0[3] = (uint32_t)((ga >> 32) & 0x1FFFFFFu) | (2u << 30);  // global_addr[56:32], type=2
    i32x8 g1;
    g1[0] = (2 << 16);                                         // data_size = 4 bytes
    g1[1] = (int)((n_dwords & 0xFFFFu) << 16);                 // tensor_dim0[15:0]
    g1[2] = (int)((n_dwords >> 16) & 0xFFFFu) | (1 << 16);     // tensor_dim0[31:16], tensor_dim1=1
    g1[3] = (int)((n_dwords & 0xFFFFu) << 16);                 // tile_dim0 = n_dwords
    g1[4] = 0;                                                 // tile_dim1=0, tile_dim2=0 (1-D)
    g1[5] = (int)n_dwords;                                     // tensor_dim0_stride[31:0]
    g1[6] = 0;
    g1[7] = 0;
    i32x4 gz4 = {0, 0, 0, 0};
    i32x8 gz8 = {0, 0, 0, 0, 0, 0, 0, 0};
    __builtin_amdgcn_tensor_load_to_lds(g0, g1, gz4, gz4, gz8, 0);
}
#endif

// ---------------- Kernel 1: projection + per-gaussian params ----------------
// params[i*8 + {0..4}] = sx, sy, ivx_scaled, ivy_scaled, amp ; depths[i] = proj_z
__global__ void proj_kernel(const float* __restrict__ V,
                            const float* __restrict__ P,
                            const float* __restrict__ pos,
                            const float* __restrict__ scales,
                            const float* __restrict__ opac,
                            float* __restrict__ params,
                            float* __restrict__ depths) {
    int i = blockIdx.x * blockDim.x + threadIdx.x;
    if (i >= N_GAUSS) return;
    // M = V @ P (4x4); uniform -> compiler scalarizes the loads
    float M[16];
    #pragma unroll
    for (int r = 0; r < 4; ++r) {
        #pragma unroll
        for (int c = 0; c < 4; ++c) {
            float s = 0.f;
            #pragma unroll
            for (int l = 0; l < 4; ++l) s += V[r * 4 + l] * P[l * 4 + c];
            M[r * 4 + c] = s;
        }
    }
    float x = pos[i * 3 + 0], y = pos[i * 3 + 1], z = pos[i * 3 + 2];
    float p0 = M[0] * x + M[1] * y + M[2]  * z + M[3];
    float p1 = M[4] * x + M[5] * y + M[6]  * z + M[7];
    float p2 = M[8] * x + M[9] * y + M[10] * z + M[11];
    float denom = p2 + 1e-6f;
    float sxv = (p0 / denom + 1.f) * (0.5f * (float)IMG_W);
    float syv = (p1 / denom + 1.f) * (0.5f * (float)IMG_H);
    float sf  = 100.f / (p2 + 1.f);
    float s2x = scales[i * 3 + 0] * sf;
    float s2y = scales[i * 3 + 1] * sf;
    // fold -0.5*log2(e) so the raster loop uses a raw exp2
    float ivx = NEG_HALF_LOG2E / (s2x * s2x + 1e-6f);
    float ivy = NEG_HALF_LOG2E / (s2y * s2y + 1e-6f);
    params[i * 8 + 0] = sxv;
    params[i * 8 + 1] = syv;
    params[i * 8 + 2] = ivx;
    params[i * 8 + 3] = ivy;
    params[i * 8 + 4] = opac[i];
    depths[i] = p2;
}

// ---------------- Kernel 2: MLP via WMMA f32 16x16x4 ----------------
// Block = 256 threads (8 waves). Block handles 16 gaussians (rows).
// Wave w computes hidden tile rows[0..15] x cols[16w..16w+15] with f32 WMMA,
// stores relu(h+b1) to LDS, then 48 threads do the 128->3 layer + sigmoid.
__global__ void mlp_kernel(const float* __restrict__ X,    // 4096 x 256
                           const float* __restrict__ W1,   // 256 x 128
                           const float* __restrict__ b1,   // 128
                           const float* __restrict__ W2,   // 128 x 3
                           const float* __restrict__ b2,   // 3
                           float* __restrict__ params) {
    __shared__ float hdd[16 * 128];
    const int g0   = blockIdx.x * 16;
    const int lane = threadIdx.x & 31;
    const int wave = threadIdx.x >> 5;   // 0..7 -> 16-column block of W1
    const int hv   = lane >> 4;          // lane half (0/1)
    const int idx  = lane & 15;
    const int col  = wave * 16 + idx;

    v8f acc0 = {};
    v8f acc1 = {};
    const float* Xr = X + (size_t)(g0 + idx) * 256;
    for (int k = 0; k < 256; k += 8) {
        // K-slice [k, k+4): A 16x4 layout -> lane half selects K pair
        int ka = k + 2 * hv;
        v2f a0, b0;
        a0.x = Xr[ka + 0];
        a0.y = Xr[ka + 1];
        b0.x = W1[(ka + 0) * 128 + col];
        b0.y = W1[(ka + 1) * 128 + col];
        acc0 = __builtin_amdgcn_wmma_f32_16x16x4_f32(
            false, a0, false, b0, (short)0, acc0, false, false);
        // K-slice [k+4, k+8): second accumulator for ILP
        int kb = k + 4 + 2 * hv;
        v2f a1, b1v;
        a1.x = Xr[kb + 0];
        a1.y = Xr[kb + 1];
        b1v.x = W1[(kb + 0) * 128 + col];
        b1v.y = W1[(kb + 1) * 128 + col];
        acc1 = __builtin_amdgcn_wmma_f32_16x16x4_f32(
            false, a1, false, b1v, (short)0, acc1, false, false);
    }
    v8f acc = acc0 + acc1;
    float bias = b1[col];
    #pragma unroll
    for (int r = 0; r < 8; ++r) {
        int m = r + 8 * hv;           // C/D layout: vgpr r, lane half -> row
        hdd[m * 128 + col] = fmaxf(acc[r] + bias, 0.f);
    }
    __syncthreads();

    int t = threadIdx.x;
    if (t < 48) {                      // 16 gaussians x 3 channels
        int g = t / 3, c = t % 3;
        float s = b2[c];
        #pragma unroll 4
        for (int k = 0; k < 128; ++k) s += hdd[g * 128 + k] * W2[k * 3 + c];
        float cv = 1.f / (1.f + __expf(-s));
        params[(size_t)(g0 + g) * 8 + 5 + c] = cv;
    }
}

// ---------------- Kernel 3: bitonic argsort by depth + gather ----------------
__global__ void sort_kernel(const float* __restrict__ depths,
                            const float* __restrict__ params,
                            float* __restrict__ sorted) {
    __shared__ float key[N_GAUSS];
    __shared__ int   val[N_GAUSS];
    const int t = threadIdx.x;   // 1024 threads
    for (int i = t; i < N_GAUSS; i += 1024) { key[i] = depths[i]; val[i] = i; }
    __syncthreads();
    for (int k = 2; k <= N_GAUSS; k <<= 1) {
        for (int j = k >> 1; j > 0; j >>= 1) {
            for (int i = t; i < N_GAUSS; i += 1024) {
                int ixj = i ^ j;
                if (ixj > i) {
                    bool up = ((i & k) == 0);
                    float ki = key[i], kj = key[ixj];
                    if (up ? (ki > kj) : (ki < kj)) {
                        key[i] = kj; key[ixj] = ki;
                        int vi = val[i]; val[i] = val[ixj]; val[ixj] = vi;
                    }
                }
            }
            __syncthreads();
        }
    }
    // gather into packed sorted array (8 floats / gaussian)
    for (int i = t; i < N_GAUSS; i += 1024) {
        int g = val[i];
        const float4* src = (const float4*)(params + (size_t)g * 8);
        float4* dst = (float4*)(sorted + (size_t)i * 8);
        dst[0] = src[0];
        dst[1] = src[1];
    }
}

// ---------------- Kernel 4: alpha compositing rasterizer ----------------
// One pixel per lane, 16x16 pixel tile per 256-thread block. Gaussians are
// staged through LDS in 32 KB chunks. With the TDM available, chunk c+1 is
// DMA'd (tensor_load_to_lds, TENSORcnt) into the other half of a 64 KB
// double buffer while all 8 waves composite chunk c.
__global__ void raster_kernel(const float* __restrict__ sorted,
                              float* __restrict__ out) {
#ifdef HAVE_TDM
    __shared__ float4 gbuf[2][RCHUNK * 2];
#else
    __shared__ float4 gbuf[1][RCHUNK * 2];
#endif
    const int lx = threadIdx.x & 15;
    const int ly = threadIdx.x >> 4;
    const int px = blockIdx.x * 16 + lx;
    const int py = blockIdx.y * 16 + ly;
    const float fx = (float)px;
    const float fy = (float)py;
    float T = 1.f, cr = 0.f, cg = 0.f, cb = 0.f;

#ifdef HAVE_TDM
    const int nchunk = N_GAUSS / RCHUNK;
    // prologue: DMA chunk 0, wave 0 waits on its TENSORcnt, barrier releases
    if (threadIdx.x < 32) {
        tdm_load_chunk(sorted, lds_byte_addr(&gbuf[0][0]), RCHUNK * 8);
        __builtin_amdgcn_s_wait_tensorcnt(0);
    }
    __syncthreads();
    for (int c = 0; c < nchunk; ++c) {
        if (threadIdx.x < 32 && (c + 1) < nchunk) {
            tdm_load_chunk(sorted + (size_t)(c + 1) * RCHUNK * 8,
                           lds_byte_addr(&gbuf[(c + 1) & 1][0]), RCHUNK * 8);
        }
        const float4* buf = &gbuf[c & 1][0];
        #pragma unroll 4
        for (int gi = 0; gi < RCHUNK; ++gi) {
            float4 p0 = buf[gi * 2 + 0];   // sx, sy, ivxs, ivys
            float4 p1 = buf[gi * 2 + 1];   // amp, r, g, b
            float dx = fx - p0.x;
            float dy = fy - p0.y;
            float e = dx * dx * p0.z + dy * dy * p0.w;  // includes -0.5*log2e
            float alpha = p1.x * fast_exp2(e);
            float w = alpha * T;
            cr += w * p1.y;
            cg += w * p1.z;
            cb += w * p1.w;
            T *= (1.f - alpha);
        }
        __syncthreads();                       // all waves done reading buf[c&1]
        if (threadIdx.x < 32) __builtin_amdgcn_s_wait_tensorcnt(0);
        __syncthreads();                       // chunk c+1 visible to all
    }
#else
    for (int c0 = 0; c0 < N_GAUSS; c0 += RCHUNK) {
        __syncthreads();
        const float4* src = (const float4*)(sorted + (size_t)c0 * 8);
        for (int i = threadIdx.x; i < RCHUNK * 2; i += 256) gbuf[0][i] = src[i];
        __syncthreads();
        #pragma unroll 4
        for (int gi = 0; gi < RCHUNK; ++gi) {
            float4 p0 = gbuf[0][gi * 2 + 0];
            float4 p1 = gbuf[0][gi * 2 + 1];
            float dx = fx - p0.x;
            float dy = fy - p0.y;
            float e = dx * dx * p0.z + dy * dy * p0.w;
            float alpha = p1.x * fast_exp2(e);
            float w = alpha * T;
            cr += w * p1.y;
            cg += w * p1.z;
            cb += w * p1.w;
            T *= (1.f - alpha);
        }
    }
#endif
    int o = (py * IMG_W + px) * 3;
    out[o + 0] = cr;
    out[o + 1] = cg;
    out[o + 2] = cb;
}

extern "C" void kernel_launch(void* const* d_in, const int* in_sizes, int n_in,
                              void* d_out, int out_size, void* d_ws, size_t ws_size,
                              hipStream_t stream) {
    const float* V   = (const float*)d_in[0];  // viewpoint_matrix 4x4
    const float* pos = (const float*)d_in[1];  // positions 4096x3
    const float* scl = (const float*)d_in[2];  // scales 4096x3
    const float* X   = (const float*)d_in[3];  // rf_features 4096x256
    const float* opa = (const float*)d_in[4];  // opacity 4096
    const float* P   = (const float*)d_in[5];  // projection 4x4
    const float* W1  = (const float*)d_in[6];  // 256x128
    const float* b1  = (const float*)d_in[7];  // 128
    const float* W2  = (const float*)d_in[8];  // 128x3
    const float* b2  = (const float*)d_in[9];  // 3
    float* out = (float*)d_out;

    float* ws     = (float*)d_ws;
    float* params = ws;                               // 4096*8 floats
    float* depths = ws + (size_t)N_GAUSS * 8;         // 4096 floats
    float* sorted = depths + N_GAUSS;                 // 4096*8 floats (16B aligned)

    proj_kernel<<<N_GAUSS / 256, 256, 0, stream>>>(V, P, pos, scl, opa, params, depths);
    mlp_kernel<<<N_GAUSS / 16, 256, 0, stream>>>(X, W1, b1, W2, b2, params);
    sort_kernel<<<1, 1024, 0, stream>>>(depths, params, sorted);
    raster_kernel<<<dim3(IMG_W / 16, IMG_H / 16), 256, 0, stream>>>(sorted, out);
}